// GCN_28552942584270
// MI455X (gfx1250) — compile-verified
//
#include <hip/hip_runtime.h>

typedef __attribute__((ext_vector_type(2))) float v2f;
typedef __attribute__((ext_vector_type(8))) float v8f;

// ---------------- utility kernels ----------------

__global__ void k_fill(float* __restrict__ p, float v, int n) {
    int i = blockIdx.x * blockDim.x + threadIdx.x;
    if (i < n) p[i] = v;
}

// ---------------- edge weight prep ----------------

// ew = (src != dst) ? attr : 0 ; deg[dst] += ew ; stash ew in w[]
__global__ void k_edge_deg(const int* __restrict__ ei, const float* __restrict__ attr,
                           float* __restrict__ w, float* __restrict__ deg, int E) {
    int e = blockIdx.x * blockDim.x + threadIdx.x;
    if (e >= E) return;
    int s = ei[e];
    int d = ei[E + e];
    float ew = (s != d) ? attr[e] : 0.0f;
    w[e] = ew;
    if (ew != 0.0f) atomicAdd(&deg[d], ew);
}

// deg -> D^{-1/2} in place
__global__ void k_dis(float* __restrict__ deg, int n) {
    int i = blockIdx.x * blockDim.x + threadIdx.x;
    if (i >= n) return;
    float dv = deg[i];
    deg[i] = (dv > 0.0f) ? rsqrtf(fmaxf(dv, 1e-30f)) : 0.0f;
}

// w[e] = -dis[dst] * ew * dis[src]
__global__ void k_edge_scale(const int* __restrict__ ei, const float* __restrict__ dis,
                             float* __restrict__ w, int E) {
    int e = blockIdx.x * blockDim.x + threadIdx.x;
    if (e >= E) return;
    int s = ei[e];
    int d = ei[E + e];
    w[e] = -dis[d] * w[e] * dis[s];
}

// ---------------- sparse propagation (scatter-add, L2-resident) ----------------
// out[dst] += scale * w[e] * h[src], vectorized float4 per thread.
// F = 4 << F4shift features per row; consecutive threads share one edge
// (src/dst/w uniform over 1<<F4shift lanes, coalesced row gather, atomics resolved in L2).
__global__ void k_spmm(const float* __restrict__ h, const int* __restrict__ ei,
                       const float* __restrict__ w, float* __restrict__ out,
                       int E, int F4shift, float scale) {
    int idx = blockIdx.x * blockDim.x + threadIdx.x;
    int total = E << F4shift;
    if (idx >= total) return;
    int e = idx >> F4shift;
    int c = idx & ((1 << F4shift) - 1);
    float ww = w[e] * scale;
    if (ww == 0.0f) return;
    int s = ei[e];
    int d = ei[E + e];
    const float4* hp = reinterpret_cast<const float4*>(h) + ((size_t)s << F4shift) + c;
    float4 hv = *hp;
    float* op = out + (((size_t)d << F4shift) << 2) + (c << 2);
    atomicAdd(op + 0, ww * hv.x);
    atomicAdd(op + 1, ww * hv.y);
    atomicAdd(op + 2, ww * hv.z);
    atomicAdd(op + 3, ww * hv.w);
}

// ---------------- fused Chebyshev GEMM via V_WMMA_F32_16X16X4_F32 ----------------
// out = relu( Tx0@(W0-W2) + Tx1@W1 + P@W2 + bias ),  P = 2*prop(Tx1)
// Block = 8 waves; wave i computes the 16x16 tile at M-tile (blockIdx.x*8+i),
// N-tile blockIdx.y. The 3 weight slices (Kd x 16, with W0-W2 pre-fused) are
// staged once per block into LDS in [col][k] order, padded to stride 132 so the
// per-lane ds_load_b64 is 8B-aligned and bank-conflict-free across the wave.
#define GW 8                 // waves per block
#define KPAD 132             // padded K stride in LDS (max Kd=128, +4)

__global__ __launch_bounds__(GW * 32) void
k_cheb_gemm(const float* __restrict__ Tx0, const float* __restrict__ Tx1,
            const float* __restrict__ P, const float* __restrict__ W,
            const float* __restrict__ bias, float* __restrict__ out,
            int M, int Kd, int N, int Mtiles) {
    __shared__ float Wl[3 * 16 * KPAD];

    int tid  = threadIdx.x;
    int nt   = blockIdx.y;

    // cooperative weight stage: Wl[t][col][k]; t=0 carries W0 - W2
    for (int idx = tid; idx < 3 * 16 * Kd; idx += GW * 32) {
        int t  = idx / (16 * Kd);
        int r  = idx - t * 16 * Kd;
        int cc = r / Kd;
        int k  = r - cc * Kd;
        int nc = nt * 16 + cc;
        float v = W[(size_t)(t * Kd + k) * N + nc];
        if (t == 0) v -= W[(size_t)(2 * Kd + k) * N + nc];
        Wl[(t * 16 + cc) * KPAD + k] = v;
    }
    __syncthreads();

    int wave = tid >> 5;
    int mt   = blockIdx.x * GW + wave;
    if (mt >= Mtiles) return;                 // wave-uniform exit

    int lane = tid & 31;
    int col  = lane & 15;
    int klo  = (lane >> 4) << 1;              // lanes 0-15: K {k,k+1}; 16-31: {k+2,k+3}
    int row  = mt * 16 + col;                 // A-matrix row for this lane
    int ncol = nt * 16 + col;                 // output column

    const float* Txs[3] = {Tx0, Tx1, P};

    v8f c;
    float bv = bias[ncol];
#pragma unroll
    for (int i = 0; i < 8; ++i) c[i] = bv;

    for (int t = 0; t < 3; ++t) {
        const float* arow = Txs[t] + (size_t)row * Kd;
        const float* brow = &Wl[(t * 16 + col) * KPAD];
        for (int k0 = 0; k0 < Kd; k0 += 4) {
            v2f a, b;
            a.x = arow[k0 + klo];
            a.y = arow[k0 + klo + 1];
            b = *reinterpret_cast<const v2f*>(brow + k0 + klo);   // ds_load_b64
            c = __builtin_amdgcn_wmma_f32_16x16x4_f32(
                    /*neg_a=*/false, a, /*neg_b=*/false, b,
                    /*c_mod=*/(short)0, c, /*reuse_a=*/false, /*reuse_b=*/false);
        }
    }

    int rbase = mt * 16 + ((lane >> 4) << 3); // C: VGPR i -> row i (+8 for hi lanes)
#pragma unroll
    for (int i = 0; i < 8; ++i) {
        float v = fmaxf(c[i], 0.0f);          // fused ReLU
        int r = rbase + i;
        if (r < M) out[(size_t)r * N + ncol] = v;
    }
}

// ---------------- pooling + classifier ----------------

__global__ void k_count(const int* __restrict__ batch, float* __restrict__ cnt, int n) {
    int i = blockIdx.x * blockDim.x + threadIdx.x;
    if (i < n) atomicAdd(&cnt[batch[i]], 1.0f);
}

// H3 = 128 features, float4 per thread
__global__ void k_pool(const float* __restrict__ h, const int* __restrict__ batch,
                       float* __restrict__ sums, int n) {
    int idx = blockIdx.x * blockDim.x + threadIdx.x;
    int total = n << 5;  // n * 128/4
    if (idx >= total) return;
    int i = idx >> 5;
    int c = idx & 31;
    int g = batch[i];
    const float4* hp = reinterpret_cast<const float4*>(h) + ((size_t)i << 5) + c;
    float4 hv = *hp;
    float* sp = sums + ((size_t)g << 7) + (c << 2);
    atomicAdd(sp + 0, hv.x);
    atomicAdd(sp + 1, hv.y);
    atomicAdd(sp + 2, hv.z);
    atomicAdd(sp + 3, hv.w);
}

// One block (128 threads) per graph: hg = sum/max(cnt,1); out = hg @ linW + linb
__global__ void k_final(const float* __restrict__ sums, const float* __restrict__ cnt,
                        const float* __restrict__ linW, const float* __restrict__ linb,
                        float* __restrict__ out, int G, int H) {
    __shared__ float hg_s[128];
    int g = blockIdx.x;
    int f = threadIdx.x;
    float c = cnt[g];
    float hg = sums[(size_t)g * H + f] / fmaxf(c, 1.0f);
    out[(size_t)G * 2 + (size_t)g * H + f] = hg;  // second tuple element (hg)
    hg_s[f] = hg;
    __syncthreads();
    if (f < 2) {
        float acc = linb[f];
        for (int j = 0; j < H; ++j) acc += hg_s[j] * linW[j * 2 + f];
        out[g * 2 + f] = acc;                      // first tuple element (logits)
    }
}

// ---------------- host-side orchestration ----------------

extern "C" void kernel_launch(void* const* d_in, const int* in_sizes, int n_in,
                              void* d_out, int out_size, void* d_ws, size_t ws_size,
                              hipStream_t stream) {
    const float* x    = (const float*)d_in[0];
    const int*   ei   = (const int*)d_in[1];   // (2, E): row0 = src, row1 = dst
    const float* attr = (const float*)d_in[2];
    const int*   bat  = (const int*)d_in[3];
    const float* W1 = (const float*)d_in[4];  const float* b1 = (const float*)d_in[5];
    const float* W2 = (const float*)d_in[6];  const float* b2 = (const float*)d_in[7];
    const float* W3 = (const float*)d_in[8];  const float* b3 = (const float*)d_in[9];
    const float* linW = (const float*)d_in[10];
    const float* linb = (const float*)d_in[11];
    float* out = (float*)d_out;

    const int F_IN = 128, H1 = 64, H2 = 64, H3 = 128, NCLS = 2;
    int Nn = in_sizes[0] / F_IN;       // 100000
    int E  = in_sizes[2];              // 1600000
    int G  = out_size / (NCLS + H3);   // 256

    float* ws = (float*)d_ws;
    size_t off = 0;
    float* dis  = ws + off; off += (size_t)Nn;            // deg then D^-1/2 (in place)
    float* w    = ws + off; off += (size_t)E;             // edge weights
    float* A    = ws + off; off += (size_t)Nn * F_IN;     // Tx1
    float* B    = ws + off; off += (size_t)Nn * F_IN;     // P = 2*prop(Tx1)
    float* C    = ws + off; off += (size_t)Nn * F_IN;     // H1, later H3
    float* D    = ws + off; off += (size_t)Nn * H2;       // H2
    float* pool = ws + off; off += (size_t)G * H3;        // graph sums
    float* cnt  = ws + off; off += (size_t)G;             // graph counts

    auto nb = [](long n, int t) { return (unsigned)((n + t - 1) / t); };

    // --- scaled Laplacian edge weights ---
    k_fill<<<nb(Nn, 256), 256, 0, stream>>>(dis, 0.0f, Nn);
    k_edge_deg<<<nb(E, 256), 256, 0, stream>>>(ei, attr, w, dis, E);
    k_dis<<<nb(Nn, 256), 256, 0, stream>>>(dis, Nn);
    k_edge_scale<<<nb(E, 256), 256, 0, stream>>>(ei, dis, w, E);

    // --- one ChebConv layer ---
    // Tx1 = prop(in); P = 2*prop(Tx1); out = relu(in@(W0-W2) + Tx1@W1 + P@W2 + b)
    auto layer = [&](const float* in, int Fin, const float* Wt, const float* bt,
                     int Nout, float* outp) {
        int sh = (Fin == 128) ? 5 : 4;              // Fin/4 = 1<<sh
        long nf = (long)Nn * Fin;
        k_fill<<<nb(nf, 256), 256, 0, stream>>>(A, 0.0f, (int)nf);
        k_fill<<<nb(nf, 256), 256, 0, stream>>>(B, 0.0f, (int)nf);
        k_spmm<<<nb((long)E << sh, 256), 256, 0, stream>>>(in, ei, w, A, E, sh, 1.0f);
        k_spmm<<<nb((long)E << sh, 256), 256, 0, stream>>>(A, ei, w, B, E, sh, 2.0f);
        int mtiles = (Nn + 15) / 16;
        dim3 g((mtiles + GW - 1) / GW, Nout / 16);
        k_cheb_gemm<<<g, GW * 32, 0, stream>>>(in, A, B, Wt, bt, outp,
                                               Nn, Fin, Nout, mtiles);
    };

    layer(x, F_IN, W1, b1, H1, C);   // 128 -> 64
    layer(C, H1,  W2, b2, H2, D);    //  64 -> 64
    layer(D, H2,  W3, b3, H3, C);    //  64 -> 128  (reuses H1's buffer)

    // --- global mean pool + linear head ---
    k_fill<<<nb((long)G * H3 + G, 256), 256, 0, stream>>>(pool, 0.0f, G * H3 + G);
    k_count<<<nb(Nn, 256), 256, 0, stream>>>(bat, cnt, Nn);
    k_pool<<<nb((long)Nn * (H3 / 4), 256), 256, 0, stream>>>(C, bat, pool, Nn);
    k_final<<<G, H3, 0, stream>>>(pool, cnt, linW, linb, out, G, H3);
}